// define_network_11304353923596
// MI455X (gfx1250) — compile-verified
//
#include <hip/hip_runtime.h>
#include <cmath>

// ---------------------------------------------------------------------------
// Problem constants (from reference): T=32768, B=1, V=539, H=100
// ---------------------------------------------------------------------------
#define TT     32768
#define V_IN   539
#define H_DIM  100

typedef __attribute__((ext_vector_type(2))) float v2f;
typedef __attribute__((ext_vector_type(8))) float v8f;

// ---------------------------------------------------------------------------
// Phase 1: pre0[t][h] = sum_v x[t][v] * Wih0[h][v]  + (bih0[h] + bhh0[h])
// FP32 WMMA 16x16x4. One wave computes a 64(M) x 16(N) strip: 4 M-subtiles
// share each B fragment. Main K loop (134 iters) is fully branch-free; the
// K tail (539 = 4*134+3) uses clamped addresses + selects, never EXEC masks.
//   A (16x4 f32): lane holds row M=lane%16, K = 2*(lane/16) (v0) and +1 (v1)
//   B (4x16 f32): lane holds col N=lane%16, same K striping
//   C/D (16x16 f32, 8 VGPRs): VGPR r, lane l -> M = r + 8*(l/16), N = l%16
// ---------------------------------------------------------------------------
__global__ __launch_bounds__(32)
void gemm_pre0_wmma(const float* __restrict__ x,
                    const float* __restrict__ Wih0,
                    const float* __restrict__ bih0,
                    const float* __restrict__ bhh0,
                    float* __restrict__ pre0)
{
    const int lane = threadIdx.x;       // 0..31, full wave (EXEC all ones)
    const int r16  = lane & 15;
    const int hi   = lane >> 4;         // 0 or 1
    const int koff = hi << 1;           // 0 or 2

    const int m0 = blockIdx.x * 64 + r16;       // first of 4 M-subtile rows
    const int n  = blockIdx.y * 16 + r16;       // output col (may be >= H_DIM)
    const bool nv = (n < H_DIM);
    const int  ncl = nv ? n : (H_DIM - 1);      // clamped: safe garbage for n>=100

    // Per-lane base pointers already include this lane's K offset.
    const float* __restrict__ xrow = x    + (size_t)m0  * V_IN + koff;
    const float* __restrict__ wrow = Wih0 + (size_t)ncl * V_IN + koff;

    v8f acc0 = {}, acc1 = {}, acc2 = {}, acc3 = {};

    // ---- main loop: k in [0, 536), all lanes fully in-bounds (k+koff+1<=535)
    #pragma unroll 2
    for (int k = 0; k < 536; k += 4) {
        v2f b;
        b.x = wrow[k];
        b.y = wrow[k + 1];
        v2f a0, a1, a2, a3;
        a0.x = xrow[k];              a0.y = xrow[k + 1];
        a1.x = xrow[k + 16 * V_IN];  a1.y = xrow[k + 16 * V_IN + 1];
        a2.x = xrow[k + 32 * V_IN];  a2.y = xrow[k + 32 * V_IN + 1];
        a3.x = xrow[k + 48 * V_IN];  a3.y = xrow[k + 48 * V_IN + 1];
        acc0 = __builtin_amdgcn_wmma_f32_16x16x4_f32(false, a0, false, b, (short)0, acc0, false, false);
        acc1 = __builtin_amdgcn_wmma_f32_16x16x4_f32(false, a1, false, b, (short)0, acc1, false, false);
        acc2 = __builtin_amdgcn_wmma_f32_16x16x4_f32(false, a2, false, b, (short)0, acc2, false, false);
        acc3 = __builtin_amdgcn_wmma_f32_16x16x4_f32(false, a3, false, b, (short)0, acc3, false, false);
    }

    // ---- K tail: absolute k = 536+koff (and +1). Element 0 always valid
    // (536 or 538 < 539); element 1 valid only for hi==0 (537). Clamp the
    // second address into bounds and select-to-zero for hi lanes (v_cndmask,
    // no EXEC divergence, no OOB access even on the last row of x).
    {
        const int k  = 536;
        const int o1 = k + (hi ? 0 : 1);        // in-bounds always
        const float zero = 0.0f;
        v2f b;
        float bw = wrow[o1];
        b.x = wrow[k];
        b.y = hi ? zero : bw;
        v2f a0, a1, a2, a3;
        float t0 = xrow[o1];
        float t1 = xrow[o1 + 16 * V_IN];
        float t2 = xrow[o1 + 32 * V_IN];
        float t3 = xrow[o1 + 48 * V_IN];
        a0.x = xrow[k];              a0.y = hi ? zero : t0;
        a1.x = xrow[k + 16 * V_IN];  a1.y = hi ? zero : t1;
        a2.x = xrow[k + 32 * V_IN];  a2.y = hi ? zero : t2;
        a3.x = xrow[k + 48 * V_IN];  a3.y = hi ? zero : t3;
        acc0 = __builtin_amdgcn_wmma_f32_16x16x4_f32(false, a0, false, b, (short)0, acc0, false, false);
        acc1 = __builtin_amdgcn_wmma_f32_16x16x4_f32(false, a1, false, b, (short)0, acc1, false, false);
        acc2 = __builtin_amdgcn_wmma_f32_16x16x4_f32(false, a2, false, b, (short)0, acc2, false, false);
        acc3 = __builtin_amdgcn_wmma_f32_16x16x4_f32(false, a3, false, b, (short)0, acc3, false, false);
    }

    // ---- store: only valid columns; bias folded in
    if (nv) {
        const float bias = bih0[n] + bhh0[n];
        const int mbase = blockIdx.x * 64 + hi * 8;
        #pragma unroll
        for (int r = 0; r < 8; ++r) {
            pre0[(size_t)(mbase      + r) * H_DIM + n] = acc0[r] + bias;
            pre0[(size_t)(mbase + 16 + r) * H_DIM + n] = acc1[r] + bias;
            pre0[(size_t)(mbase + 32 + r) * H_DIM + n] = acc2[r] + bias;
            pre0[(size_t)(mbase + 48 + r) * H_DIM + n] = acc3[r] + bias;
        }
    }
}

// ---------------------------------------------------------------------------
// Phase 2: sequential 2-layer Elman scan + output heads.
// Single workgroup. All three recurrent weight matrices (120 KB) live in the
// WGP's 320 KB LDS, stored COLUMN-major so thread-i-per-output GEMV reads
// consecutive LDS addresses across the wave (bank-conflict free).
//   h0 = tanh(pre0[t] + Whh0 @ h0)
//   h1 = tanh(Wih1 @ h0 + (bih1+bhh1) + Whh1 @ h1)
// ---------------------------------------------------------------------------
__global__ __launch_bounds__(128)
void rnn_scan_heads(const float* __restrict__ pre0,
                    const float* __restrict__ Whh0,
                    const float* __restrict__ Wih1,
                    const float* __restrict__ Whh1,
                    const float* __restrict__ bih1,
                    const float* __restrict__ bhh1,
                    const float* __restrict__ Wc,   const float* __restrict__ bc,
                    const float* __restrict__ Wcol, const float* __restrict__ bcol,
                    const float* __restrict__ Wg,   const float* __restrict__ bg,
                    const float* __restrict__ Ws,   const float* __restrict__ bs,
                    float* __restrict__ out)
{
    extern __shared__ float smem[];
    float* w0  = smem;            // [100*100] Whh0, column-major
    float* w1i = w0  + 10000;     // [100*100] Wih1, column-major
    float* w1h = w1i + 10000;     // [100*100] Whh1, column-major
    float* h0s = w1h + 10000;     // [100] (+pad)
    float* h1s = h0s + 104;       // [100]
    float* b1s = h1s + 104;       // [100] bih1 + bhh1

    const int tid = threadIdx.x;

    // Transpose-load weights into LDS: cm[j*100 + i] = W[i*100 + j]
    for (int idx = tid; idx < H_DIM * H_DIM; idx += blockDim.x) {
        const int i = idx % H_DIM;
        const int j = idx / H_DIM;
        const int src = i * H_DIM + j;
        w0[idx]  = Whh0[src];
        w1i[idx] = Wih1[src];
        w1h[idx] = Whh1[src];
    }
    if (tid < H_DIM) {
        h0s[tid] = 0.0f;
        h1s[tid] = 0.0f;
        b1s[tid] = bih1[tid] + bhh1[tid];
    }
    __syncthreads();

    for (int t = 0; t < TT; ++t) {
        float s0 = 0.0f, s1 = 0.0f;
        if (tid < H_DIM) {
            s0 = pre0[(size_t)t * H_DIM + tid];   // already includes bih0+bhh0
            s1 = b1s[tid];
            #pragma unroll 4
            for (int j = 0; j < H_DIM; ++j) {
                s0 += w0[j * H_DIM + tid]  * h0s[j];   // Whh0 @ h0   (old)
                s1 += w1h[j * H_DIM + tid] * h1s[j];   // Whh1 @ h1   (old)
            }
            s0 = tanhf(s0);
        }
        __syncthreads();                  // everyone done reading old h0s
        if (tid < H_DIM) h0s[tid] = s0;   // publish new h0
        __syncthreads();
        if (tid < H_DIM) {
            #pragma unroll 4
            for (int j = 0; j < H_DIM; ++j)
                s1 += w1i[j * H_DIM + tid] * h0s[j];   // Wih1 @ h0 (new)
            s1 = tanhf(s1);
        }
        __syncthreads();                  // everyone done reading old h1s
        if (tid < H_DIM) h1s[tid] = s1;   // publish new h1
        __syncthreads();
    }

    // Outputs, flat in return order: hn2[100], cate[19], color[17], gender[2], sleeve[4]
    if (tid < H_DIM) out[tid] = h1s[tid];
    if (tid < 42) {
        const float* W; const float* b; int o, i;
        if (tid < 19)      { W = Wc;   b = bc;   o = 100; i = tid;      }
        else if (tid < 36) { W = Wcol; b = bcol; o = 119; i = tid - 19; }
        else if (tid < 38) { W = Wg;   b = bg;   o = 136; i = tid - 36; }
        else               { W = Ws;   b = bs;   o = 138; i = tid - 38; }
        float s = b[i];
        #pragma unroll 4
        for (int j = 0; j < H_DIM; ++j) s += W[i * H_DIM + j] * h1s[j];
        out[o + i] = s;
    }
}

// ---------------------------------------------------------------------------
extern "C" void kernel_launch(void* const* d_in, const int* in_sizes, int n_in,
                              void* d_out, int out_size, void* d_ws, size_t ws_size,
                              hipStream_t stream) {
    const float* x    = (const float*)d_in[0];
    const float* Wih0 = (const float*)d_in[1];
    const float* Whh0 = (const float*)d_in[2];
    const float* bih0 = (const float*)d_in[3];
    const float* bhh0 = (const float*)d_in[4];
    const float* Wih1 = (const float*)d_in[5];
    const float* Whh1 = (const float*)d_in[6];
    const float* bih1 = (const float*)d_in[7];
    const float* bhh1 = (const float*)d_in[8];
    const float* Wc   = (const float*)d_in[9];
    const float* bc   = (const float*)d_in[10];
    const float* Wcol = (const float*)d_in[11];
    const float* bcol = (const float*)d_in[12];
    const float* Wg   = (const float*)d_in[13];
    const float* bg   = (const float*)d_in[14];
    const float* Ws   = (const float*)d_in[15];
    const float* bs   = (const float*)d_in[16];

    float* out  = (float*)d_out;
    float* pre0 = (float*)d_ws;            // TT * H_DIM floats = 13.1 MB

    // Phase 1: FP32 WMMA GEMM, 512 x 7 blocks, each wave does a 64x16 strip.
    dim3 grid(TT / 64, (H_DIM + 15) / 16, 1);
    gemm_pre0_wmma<<<grid, 32, 0, stream>>>(x, Wih0, bih0, bhh0, pre0);

    // Phase 2: single-workgroup sequential scan, 121 KB dynamic LDS.
    const size_t smemBytes = (size_t)(3 * 10000 + 3 * 104) * sizeof(float);
    (void)hipFuncSetAttribute((const void*)rnn_scan_heads,
                              hipFuncAttributeMaxDynamicSharedMemorySize,
                              (int)smemBytes);
    rnn_scan_heads<<<1, 128, smemBytes, stream>>>(
        pre0, Whh0, Wih1, Whh1, bih1, bhh1,
        Wc, bc, Wcol, bcol, Wg, bg, Ws, bs, out);
}